// BilateralFilter_55791625175588
// MI455X (gfx1250) — compile-verified
//
#include <hip/hip_runtime.h>
#include <hip/hip_bf16.h>

// ---------------------------------------------------------------------------
// Bilateral 5x5 filter, (B=4,C=3,H=512,W=512) f32, reflect padding.
// Transcendental-bound stencil: 25 v_exp_f32 per pixel. WMMA does not apply
// (weights are data-dependent -> per-pixel dot, not a matmul). We instead use
// the CDNA5 async global->LDS path to stage tiles, ds_load_b128 for row reads,
// and one fused exp2 per tap.
// ---------------------------------------------------------------------------

#define IMG      512
#define KSZ      5
#define PAD      2
#define TILE_W   64
#define TILE_H   16
#define LW       (TILE_W + 2 * PAD)   // 68 (multiple of 4 -> 16B-aligned rows)
#define LH       (TILE_H + 2 * PAD)   // 20
#define LDS_N    (LW * LH)            // 1360 floats = 5440 B

// exp(-d^2/(2*0.1^2)) == exp2(d^2 * (-50 * log2(e)))
#define RANGE_C2 (-72.134752044448169f)
#define EPS      1e-8f

#if defined(__AMDGCN__) && defined(__has_builtin)
#  if __has_builtin(__builtin_amdgcn_global_load_async_to_lds_b32)
#    define HAS_ASYNC_LDS 1
#  endif
#  if __has_builtin(__builtin_amdgcn_s_wait_asynccnt)
#    define HAS_WAIT_ASYNC_BUILTIN 1
#  endif
#endif
#ifndef HAS_ASYNC_LDS
#  define HAS_ASYNC_LDS 0
#endif

#if defined(__AMDGCN__)
typedef __attribute__((address_space(1))) int g_int_t;   // global
typedef __attribute__((address_space(3))) int l_int_t;   // LDS
#endif

__device__ __forceinline__ int reflect_idx(int g) {
    // jnp 'reflect' (no edge repeat): -1 -> 1, N -> N-2
    g = (g < 0) ? -g : g;
    g = (g >= IMG) ? (2 * IMG - 2 - g) : g;
    return g;
}

__global__ __launch_bounds__(256)
void bilateral5x5_kernel(const float* __restrict__ x,
                         const float* __restrict__ sk,   // 25 spatial weights
                         float* __restrict__ out) {
    __shared__ float tile[LDS_N];

    const int tid   = threadIdx.x;
    const int plane = blockIdx.z;                       // B*C planes
    const int ty0   = blockIdx.y * TILE_H;
    const int tx0   = blockIdx.x * TILE_W;

    const float* __restrict__ src = x + (size_t)plane * IMG * IMG;

    // ---- Stage (68 x 20) reflected tile into LDS (async global->LDS) ----
    for (int i = tid; i < LDS_N; i += 256) {
        const int lr = i / LW;
        const int lc = i - lr * LW;
        const int gy = reflect_idx(ty0 - PAD + lr);
        const int gx = reflect_idx(tx0 - PAD + lc);
        const float* gp = src + gy * IMG + gx;
#if HAS_ASYNC_LDS
        __builtin_amdgcn_global_load_async_to_lds_b32(
            (g_int_t*)(size_t)gp,
            (l_int_t*)&tile[i],
            /*offset=*/0, /*cpol=*/0);
#else
        tile[i] = *gp;
#endif
    }
#if HAS_ASYNC_LDS
#  if defined(HAS_WAIT_ASYNC_BUILTIN)
    __builtin_amdgcn_s_wait_asynccnt(0);
#  else
    asm volatile("s_wait_asynccnt 0x0" ::: "memory");
#  endif
#endif
    __syncthreads();

    // ---- Spatial weights (uniform -> scalar loads) ----
    float w[KSZ * KSZ];
#pragma unroll
    for (int t = 0; t < KSZ * KSZ; ++t) w[t] = sk[t];

    // ---- Each thread: 4 horizontally adjacent pixels ----
    const int tx = (tid & 15) * 4;  // 0..60, multiple of 4
    const int ty = tid >> 4;        // 0..15

    float ctr[4];
#pragma unroll
    for (int j = 0; j < 4; ++j)
        ctr[j] = tile[(ty + PAD) * LW + tx + PAD + j];

    float acc[4] = {0.f, 0.f, 0.f, 0.f};
    float wsm[4] = {0.f, 0.f, 0.f, 0.f};

#pragma unroll
    for (int dy = 0; dy < KSZ; ++dy) {
        // Row segment [tx .. tx+7] of padded tile: two ds_load_b128
        const float4* rp = (const float4*)&tile[(ty + dy) * LW + tx];
        const float4 a = rp[0];
        const float4 b = rp[1];
        const float r[8] = {a.x, a.y, a.z, a.w, b.x, b.y, b.z, b.w};
#pragma unroll
        for (int dj = 0; dj < KSZ; ++dj) {
            const float s = w[dy * KSZ + dj];
#pragma unroll
            for (int j = 0; j < 4; ++j) {
                const float p = r[dj + j];
                const float d = p - ctr[j];
#if defined(__AMDGCN__)
                const float e = __builtin_amdgcn_exp2f(d * d * RANGE_C2);
#else
                const float e = exp2f(d * d * RANGE_C2);
#endif
                const float wt = e * s;
                wsm[j] += wt;
                acc[j] = fmaf(wt, p, acc[j]);
            }
        }
    }

    float4 o;
#if defined(__AMDGCN__)
    o.x = acc[0] * __builtin_amdgcn_rcpf(wsm[0] + EPS);
    o.y = acc[1] * __builtin_amdgcn_rcpf(wsm[1] + EPS);
    o.z = acc[2] * __builtin_amdgcn_rcpf(wsm[2] + EPS);
    o.w = acc[3] * __builtin_amdgcn_rcpf(wsm[3] + EPS);
#else
    o.x = acc[0] / (wsm[0] + EPS);
    o.y = acc[1] / (wsm[1] + EPS);
    o.z = acc[2] / (wsm[2] + EPS);
    o.w = acc[3] / (wsm[3] + EPS);
#endif

    float* op = out + (size_t)plane * IMG * IMG + (size_t)(ty0 + ty) * IMG + tx0 + tx;
    *(float4*)op = o;
}

extern "C" void kernel_launch(void* const* d_in, const int* in_sizes, int n_in,
                              void* d_out, int out_size, void* d_ws, size_t ws_size,
                              hipStream_t stream) {
    const float* x  = (const float*)d_in[0];   // (B, C, 512, 512) f32
    const float* sk = (const float*)d_in[1];   // (5, 5) f32
    float* out      = (float*)d_out;

    const int planes = in_sizes[0] / (IMG * IMG);   // B*C = 12
    dim3 grid(IMG / TILE_W, IMG / TILE_H, planes);  // (8, 32, 12)
    bilateral5x5_kernel<<<grid, 256, 0, stream>>>(x, sk, out);
}